// YOLO_28054726377592
// MI455X (gfx1250) — compile-verified
//
#include <hip/hip_runtime.h>
#include <stdint.h>

// ---------------- problem constants ----------------
#define NCLS    80
#define BATCH   16
#define ANC     3
#define DIMG    26
#define CELLS   (DIMG*DIMG)          // 676
#define INST    (BATCH*ANC)          // 48
#define PLANE   (INST*CELLS)         // 32448  (B*3*dim*dim)
#define TGT     20
#define SLOTS   (BATCH*TGT)          // 320
#define RSTR    8                    // padded row stride (7 used)
#define ROWS_FLOATS  (INST*CELLS*RSTR)
#define PREDS_FLOATS (PLANE*7)
#define NMS_TH  0.4f
#define IGN_TH  0.5f
#define PAD     1024                 // bitonic size >= 676

// anchors (reversed list, NUM=0 -> first three), divided by stride 416/26=16
__device__ __constant__ float d_aw[3] = {373.0f/16.0f, 156.0f/16.0f, 116.0f/16.0f};
__device__ __constant__ float d_ah[3] = {326.0f/16.0f, 198.0f/16.0f,  90.0f/16.0f};

__device__ __forceinline__ float sigm(float v) { return 1.0f / (1.0f + __expf(-v)); }

// ---------------- TDM helper: 1-D tile Global -> LDS ----------------
typedef unsigned int u32x4 __attribute__((ext_vector_type(4)));
typedef int          i32x4 __attribute__((ext_vector_type(4)));
typedef int          i32x8 __attribute__((ext_vector_type(8)));

__device__ __forceinline__ void tdm_load_1d(const void* gptr, void* lptr, unsigned elems /*4B units*/) {
  unsigned long long ga = (unsigned long long)(uintptr_t)gptr;
  unsigned lds = (unsigned)(uintptr_t)lptr;   // low 32 bits of generic LDS ptr = LDS byte offset
  // D# group0: count=1 | lds_addr | global_addr[31:0] | global_addr[56:32], type=2 ("image")
  u32x4 g0 = { 1u,
               lds,
               (unsigned)ga,
               ((unsigned)((ga >> 32) & 0x1FFFFFFull)) | 0x80000000u };
  // D# group1: data_size=2 (4B); tensor_dim0=elems; tensor_dim1=1; tile_dim0=elems;
  //            tile_dim1/2=0 (unused); tensor_dim0_stride=elems
  i32x8 g1 = { (int)(2u << 16),
               (int)((elems & 0xFFFFu) << 16),
               (int)((elems >> 16) | (1u << 16)),
               (int)((elems & 0xFFFFu) << 16),
               0,
               (int)elems,
               0, 0 };
  i32x4 z4 = {0, 0, 0, 0};
#if defined(__clang_major__) && (__clang_major__ >= 23)
  i32x8 z8 = {0, 0, 0, 0, 0, 0, 0, 0};
  __builtin_amdgcn_tensor_load_to_lds(g0, g1, z4, z4, z8, 0);
#else
  __builtin_amdgcn_tensor_load_to_lds(g0, g1, z4, z4, 0);
#endif
}

// ---------------- kernel 0: init outputs / scratch ----------------
__global__ void k_init(float* __restrict__ out, float* __restrict__ preds) {
  const size_t ZBASE = 85ull * PLANE;            // tx starts here
  const size_t ZSPAN = 87ull * PLANE + 2ull;     // through conf_mask + nGT + nOK
  const size_t C1LO  = 171ull * PLANE;           // conf_mask region: init to 1.0
  const size_t C1HI  = 172ull * PLANE;
  const size_t TOTAL = ZSPAN + (size_t)PREDS_FLOATS;
  for (size_t i = (size_t)blockIdx.x * blockDim.x + threadIdx.x; i < TOTAL;
       i += (size_t)gridDim.x * blockDim.x) {
    if (i < ZSPAN) {
      size_t oi = ZBASE + i;
      out[oi] = (oi >= C1LO && oi < C1HI) ? 1.0f : 0.0f;
    } else {
      preds[i - ZSPAN] = 0.0f;
    }
  }
}

// ---------------- kernel 1: activations + NMS rows ----------------
__global__ void k_prep(const float* __restrict__ x, float* __restrict__ out,
                       float* __restrict__ rows) {
  const int inst = blockIdx.y;                    // b*3 + a
  const int a = inst % ANC;
  const int b = inst / ANC;
  const int cell = blockIdx.x * blockDim.x + threadIdx.x;
  if (cell >= CELLS) return;
  const float* xb = x + ((size_t)b * 255 + (size_t)a * 85) * CELLS;

  const int gi = cell / DIMG, gj = cell % DIMG;
  const float v0 = xb[0 * CELLS + cell];
  const float v1 = xb[1 * CELLS + cell];
  const float v2 = xb[2 * CELLS + cell];
  const float v3 = xb[3 * CELLS + cell];
  const float v4 = xb[4 * CELLS + cell];
  const float px = sigm(v0), py = sigm(v1), pc = sigm(v4);

  const size_t o = (size_t)inst * CELLS + cell;
  out[o]                 = px;
  out[1ull * PLANE + o]  = py;
  out[2ull * PLANE + o]  = v2;   // pw (raw)
  out[3ull * PLANE + o]  = v3;   // ph (raw)
  out[4ull * PLANE + o]  = pc;

  float best = -1.0f; int bi = 0;
  float* ppo = out + 5ull * PLANE + o * NCLS;
  for (int c = 0; c < NCLS; ++c) {
    float p = sigm(xb[(5 + c) * CELLS + cell]);
    ppo[c] = p;
    if (p > best) { best = p; bi = c; }          // first-max (argmax semantics)
  }

  const float sx = px + (float)gi;
  const float sy = py + (float)gj;
  const float sw = __expf(v2) * d_aw[a];
  const float sh = __expf(v3) * d_ah[a];
  const float lim = (float)(DIMG - 1);
  const float xmin = fminf(fmaxf(sx - 0.5f * sw, 0.0f), lim);
  const float xmax = fminf(fmaxf(sx + 0.5f * sw, 0.0f), lim);
  const float ymin = fminf(fmaxf(sy - 0.5f * sh, 0.0f), lim);
  const float ymax = fminf(fmaxf(sy + 0.5f * sh, 0.0f), lim);

  float* r = rows + ((size_t)inst * CELLS + cell) * RSTR;
  r[0] = xmin; r[1] = xmax; r[2] = ymin; r[3] = ymax;   // reference's stacking order
  r[4] = pc;   r[5] = (float)bi; r[6] = best; r[7] = 0.0f;
}

// ---------------- kernel 2: per-instance NMS + preds scatter ----------------
__global__ void __launch_bounds__(512) k_nms(const float* __restrict__ rows,
                                             float* __restrict__ preds) {
  __shared__ float skey[PAD];
  __shared__ int   sidx[PAD];
  __shared__ float srows[CELLS * RSTR];   // 21632 B staged via TDM
  __shared__ int   skeep[CELLS];

  const int inst = blockIdx.x;
  const int tid = threadIdx.x;
  const float* grows = rows + (size_t)inst * CELLS * RSTR;

  // wave 0 issues the async tensor DMA (EXEC-independent) and waits on TENSORcnt;
  // remaining waves proceed to key init in parallel.
  if (tid == 0) {
    tdm_load_1d(grows, srows, CELLS * RSTR);
    __builtin_amdgcn_s_wait_tensorcnt(0);
  }

  // sort keys straight from global (overlaps with TDM stream into LDS)
  for (int i = tid; i < PAD; i += blockDim.x) {
    if (i < CELLS) { skey[i] = grows[i * RSTR + 4]; sidx[i] = i; skeep[i] = 1; }
    else           { skey[i] = -__builtin_inff();  sidx[i] = i; }
  }
  __syncthreads();

  // stable bitonic sort: score descending, original index ascending on ties
  for (int k = 2; k <= PAD; k <<= 1) {
    for (int j = k >> 1; j > 0; j >>= 1) {
      for (int e = tid; e < PAD; e += blockDim.x) {
        int p = e ^ j;
        if (p > e) {
          float k0 = skey[e], k1 = skey[p];
          int   i0 = sidx[e], i1 = sidx[p];
          bool before = (k0 > k1) || (k0 == k1 && i0 < i1);
          bool up = ((e & k) == 0);
          if (up ? !before : before) {
            skey[e] = k1; skey[p] = k0;
            sidx[e] = i1; sidx[p] = i0;
          }
        }
      }
      __syncthreads();
    }
  }

  // sequential suppression scan (order-exact vs. reference lax.scan)
  for (int i = 0; i < CELLS; ++i) {
    if (skeep[i]) {
      const float* ri = &srows[sidx[i] * RSTR];
      const float r0 = ri[0], r1 = ri[1], r2 = ri[2], r3 = ri[3], rc = ri[5];
      const float a1 = (r2 - r0 + 1.0f) * (r3 - r1 + 1.0f);
      for (int jj = i + 1 + tid; jj < CELLS; jj += blockDim.x) {
        if (!skeep[jj]) continue;
        const float* rj = &srows[sidx[jj] * RSTR];
        if (rj[5] != rc) continue;
        const float a2 = (rj[2] - rj[0] + 1.0f) * (rj[3] - rj[1] + 1.0f);
        const float x0 = fmaxf(r0, rj[0]), y0 = fmaxf(r1, rj[1]);
        const float x1 = fminf(r2, rj[2]), y1 = fminf(r3, rj[3]);
        const float inter = fmaxf(x1 - x0, 0.0f) * fmaxf(y1 - y0, 0.0f);
        const float iou = fmaxf(inter / (a1 + a2 - inter), 0.0f);
        if (iou >= NMS_TH) skeep[jj] = 0;
      }
    }
    __syncthreads();
  }

  // scatter kept boxes into preds; ascending rank, last write wins (dup resolution)
  if (tid == 0) {
    float* pb = preds + (size_t)inst * CELLS * 7;
    for (int r = 0; r < CELLS; ++r) {
      if (!skeep[r]) continue;                     // xi==dim -> dropped
      const float* rr = &srows[sidx[r] * RSTR];
      const float c1 = 0.5f * (rr[0] + rr[2]);
      const float c2 = 0.5f * (rr[1] + rr[3]);
      const int xi = (int)c1, yi = (int)c2;
      if (xi < 0 || xi >= DIMG || yi < 0 || yi >= DIMG) continue;
      float* q = pb + ((size_t)xi * DIMG + yi) * 7;
      q[0] = c1;          q[1] = c2;
      q[2] = rr[2] - c1;  q[3] = rr[3] - c2;
      q[4] = rr[4];       q[5] = rr[5];      q[6] = rr[6];
    }
  }
}

// ---------------- kernel 3: build_targets ----------------
__global__ void __launch_bounds__(512) k_targets(const float* __restrict__ targets,
                                                 const float* __restrict__ preds,
                                                 float* __restrict__ out) {
  __shared__ float s_tx[SLOTS], s_ty[SLOTS], s_tw[SLOTS], s_th[SLOTS];
  __shared__ float s_txv[SLOTS], s_tyv[SLOTS], s_twv[SLOTS], s_thv[SLOTS];
  __shared__ int   s_cls[SLOTS], s_xC[SLOTS], s_yC[SLOTS], s_aN[SLOTS];
  __shared__ int   s_val[SLOTS], s_ign[SLOTS];
  __shared__ int   s_cnt[2];

  const int tid = threadIdx.x;
  if (tid < 2) s_cnt[tid] = 0;
  __syncthreads();

  if (tid < SLOTS) {
    const float* tr = targets + (size_t)tid * 5;
    const float sum = tr[0] + tr[1] + tr[2] + tr[3] + tr[4];
    const int v = (sum != 0.0f);
    const float txs = tr[1] * DIMG, tys = tr[2] * DIMG;
    const float tws = tr[3] * DIMG, ths = tr[4] * DIMG;
    const int cls = (int)tr[0];
    const int xC = (int)txs, yC = (int)tys;
    const float a1 = (tws + 1.0f) * (ths + 1.0f);
    float best = -1.0e30f; int an = 0; int ign = 0;
    for (int a = 0; a < ANC; ++a) {
      const float a2 = (d_aw[a] + 1.0f) * (d_ah[a] + 1.0f);
      const float inter = fmaxf(fminf(tws, d_aw[a]), 0.0f) * fmaxf(fminf(ths, d_ah[a]), 0.0f);
      const float iou = fmaxf(inter / (a1 + a2 - inter), 0.0f);
      if (iou > best) { best = iou; an = a; }
      if (iou > IGN_TH) ign |= (1 << a);
    }
    s_val[tid] = v;  s_cls[tid] = cls;  s_xC[tid] = xC;  s_yC[tid] = yC;
    s_aN[tid] = an;  s_ign[tid] = v ? ign : 0;
    s_tx[tid] = txs; s_ty[tid] = tys; s_tw[tid] = tws; s_th[tid] = ths;
    s_txv[tid] = txs - (float)xC;
    s_tyv[tid] = tys - (float)yC;
    s_twv[tid] = __logf(tws / d_aw[an] + 1e-16f);
    s_thv[tid] = __logf(ths / d_ah[an] + 1e-16f);
    if (v) atomicAdd(&s_cnt[0], 1);
  }
  __syncthreads();

  float* o_tx    = out + 85ull  * PLANE;
  float* o_ty    = out + 86ull  * PLANE;
  float* o_tw    = out + 87ull  * PLANE;
  float* o_th    = out + 88ull  * PLANE;
  float* o_tconf = out + 89ull  * PLANE;
  float* o_tcls  = out + 90ull  * PLANE;
  float* o_mask  = out + 170ull * PLANE;
  float* o_conf  = out + 171ull * PLANE;

  // exact sequential scan over targets (plane conf-zero, then scatter-sets)
  for (int t = 0; t < TGT; ++t) {
    for (int w = tid; w < BATCH * ANC * CELLS; w += blockDim.x) {
      const int cell = w % CELLS;
      const int ba = w / CELLS;
      const int a = ba % ANC, b = ba / ANC;
      if (s_ign[b * TGT + t] & (1 << a)) o_conf[(size_t)ba * CELLS + cell] = 0.0f;
    }
    __threadfence_block();
    __syncthreads();
    if (tid < BATCH) {
      const int slot = tid * TGT + t;
      const int v = s_val[slot];
      const int a = s_aN[slot];
      const int xs = v ? s_xC[slot] : DIMG;      // invalid -> OOB -> dropped
      const int ys = s_yC[slot];
      if (xs >= 0 && xs < DIMG && ys >= 0 && ys < DIMG) {
        const size_t idx = (((size_t)tid * ANC + a) * DIMG + xs) * DIMG + ys;
        o_mask[idx] = 1.0f;  o_conf[idx] = 1.0f;
        o_tx[idx] = s_txv[slot]; o_ty[idx] = s_tyv[slot];
        o_tw[idx] = s_twv[slot]; o_th[idx] = s_thv[slot];
        o_tconf[idx] = 1.0f;
        const int c = s_cls[slot];
        if (c >= 0 && c < NCLS) o_tcls[idx * NCLS + c] = 1.0f;
      }
    }
    __threadfence_block();
    __syncthreads();
  }

  // nOK: IoU(target box, predicted box at matched cell) > 0.5
  if (tid < SLOTS && s_val[tid]) {
    const int b = tid / TGT;
    const int xg = min(max(s_xC[tid], 0), DIMG - 1);
    const int yg = min(max(s_yC[tid], 0), DIMG - 1);
    const float* q = preds + ((((size_t)b * ANC + s_aN[tid]) * DIMG + xg) * DIMG + yg) * 7;
    const float p0 = q[0], p1 = q[1], p2 = q[2], p3 = q[3];
    const float t0 = s_tx[tid] - 0.5f * s_tw[tid], t1 = s_ty[tid] - 0.5f * s_th[tid];
    const float t2 = s_tx[tid] + 0.5f * s_tw[tid], t3 = s_ty[tid] + 0.5f * s_th[tid];
    const float u0 = p0 - 0.5f * p2, u1 = p1 - 0.5f * p3;
    const float u2 = p0 + 0.5f * p2, u3 = p1 + 0.5f * p3;
    const float A1 = (t2 - t0 + 1.0f) * (t3 - t1 + 1.0f);
    const float A2 = (u2 - u0 + 1.0f) * (u3 - u1 + 1.0f);
    const float x0 = fmaxf(t0, u0), y0 = fmaxf(t1, u1);
    const float x1 = fminf(t2, u2), y1 = fminf(t3, u3);
    const float inter = fmaxf(x1 - x0, 0.0f) * fmaxf(y1 - y0, 0.0f);
    const float iou = fmaxf(inter / (A1 + A2 - inter), 0.0f);
    if (iou > 0.5f) atomicAdd(&s_cnt[1], 1);
  }
  __syncthreads();
  if (tid == 0) {
    out[172ull * PLANE]     = (float)s_cnt[0];   // nGT
    out[172ull * PLANE + 1] = (float)s_cnt[1];   // nOK
  }
}

// ---------------- host side ----------------
extern "C" void kernel_launch(void* const* d_in, const int* in_sizes, int n_in,
                              void* d_out, int out_size, void* d_ws, size_t ws_size,
                              hipStream_t stream) {
  (void)in_sizes; (void)n_in; (void)out_size; (void)ws_size;
  const float* x  = (const float*)d_in[0];   // (16, 255, 26, 26) f32
  const float* tg = (const float*)d_in[1];   // (16, 20, 5) f32
  float* out   = (float*)d_out;
  float* rows  = (float*)d_ws;               // 48*676*8 f32 NMS rows
  float* preds = rows + ROWS_FLOATS;         // (16,3,26,26,7) f32

  k_init<<<2048, 256, 0, stream>>>(out, preds);
  dim3 pg((CELLS + 255) / 256, INST);
  k_prep<<<pg, 256, 0, stream>>>(x, out, rows);
  k_nms<<<INST, 512, 0, stream>>>(rows, preds);
  k_targets<<<1, 512, 0, stream>>>(tg, preds, out);
}